// Wav2vec2Loss_15582141350648
// MI455X (gfx1250) — compile-verified
//
#include <hip/hip_runtime.h>
#include <hip/hip_bf16.h>
#include <stdint.h>

// ---- static problem config (matches reference) ----
#define B_   16
#define T_   512
#define D_   256
#define M_   256
#define K_   100
#define N_   (B_*M_)          // 4096
#define G_   2
#define V_   320
#define TEMP_INV 10.0f        // 1/0.1
#define ALPHA_   0.1f
#define COS_EPS_ 1e-8f

typedef __attribute__((ext_vector_type(2))) float v2f;
typedef __attribute__((ext_vector_type(8))) float v8f;

// ---------------------------------------------------------------------------
// 1) per-batch-row prefix scan over the boolean mask -> masked frame positions
// ---------------------------------------------------------------------------
__global__ void k_positions(const unsigned char* __restrict__ mask,
                            int* __restrict__ pos) {
  __shared__ int s[T_];
  int b = blockIdx.x, t = threadIdx.x;
  int m = mask[b * T_ + t] ? 1 : 0;
  s[t] = m;
  __syncthreads();
  // Hillis-Steele inclusive scan over 512 entries
  for (int off = 1; off < T_; off <<= 1) {
    int v = (t >= off) ? s[t - off] : 0;
    __syncthreads();
    s[t] += v;
    __syncthreads();
  }
  if (m) pos[b * M_ + (s[t] - 1)] = t;   // exactly M_ trues per row by construction
}

// ---------------------------------------------------------------------------
// 2) gather masked rows + cosine-normalize (x / max(||x||, eps))
// ---------------------------------------------------------------------------
__global__ void k_gather_norm(const float* __restrict__ enc,
                              const float* __restrict__ quant,
                              const int* __restrict__ pos,
                              float* __restrict__ t_hat,
                              float* __restrict__ l_hat) {
  __shared__ float red[D_];
  int n = blockIdx.x;          // target row 0..N-1
  int d = threadIdx.x;         // 0..255
  int b = n / M_;
  int f = pos[n];
  size_t src = ((size_t)(b * T_ + f)) * D_ + d;
  float te = enc[src];
  float le = quant[src];

  red[d] = te * te; __syncthreads();
  for (int off = 128; off > 0; off >>= 1) { if (d < off) red[d] += red[d + off]; __syncthreads(); }
  float tscale = 1.0f / fmaxf(sqrtf(red[0]), COS_EPS_);
  __syncthreads();

  red[d] = le * le; __syncthreads();
  for (int off = 128; off > 0; off >>= 1) { if (d < off) red[d] += red[d + off]; __syncthreads(); }
  float lscale = 1.0f / fmaxf(sqrtf(red[0]), COS_EPS_);

  t_hat[(size_t)n * D_ + d] = te * tscale;
  l_hat[(size_t)n * D_ + d] = le * lscale;
}

// ---------------------------------------------------------------------------
// 3) negatives: JAX threefry2x32(key=42) uniforms + stable argsort (bitonic on
//    u64 (float_bits<<32)|index keys), skip-self remap -> global label indices
// ---------------------------------------------------------------------------
__device__ inline uint2 tf2x32(uint32_t x0, uint32_t x1) {
  const uint32_t k0 = 0u, k1 = 42u;                      // jax.random.key(42)
  uint32_t ks[3] = { k0, k1, k0 ^ k1 ^ 0x1BD11BDAu };
  uint32_t X0 = x0 + ks[0], X1 = x1 + ks[1];
  const int r0[4] = {13, 15, 26, 6};
  const int r1[4] = {17, 29, 16, 24};
#pragma unroll
  for (int i = 0; i < 5; ++i) {
    const int* rr = (i & 1) ? r1 : r0;
#pragma unroll
    for (int j = 0; j < 4; ++j) {
      X0 += X1;
      int r = rr[j];
      X1 = (X1 << r) | (X1 >> (32 - r));
      X1 ^= X0;
    }
    X0 += ks[(i + 1) % 3];
    X1 += ks[(i + 2) % 3] + (uint32_t)(i + 1);
  }
  uint2 y; y.x = X0; y.y = X1; return y;
}

__global__ void k_negatives(int* __restrict__ neg_idx) {
  __shared__ unsigned long long keys[256];
  int n = blockIdx.x;
  int e = threadIdx.x;                 // candidate index 0..255 (255 = pad)
  const long long SIZE = (long long)N_ * (M_ - 1);   // 1,044,480 (even)
  const long long H = SIZE / 2;

  unsigned long long kv = 0xFFFFFFFFFFFFFFFFull;
  if (e < M_ - 1) {
    long long p = (long long)n * (M_ - 1) + e;       // global iota position
    uint32_t bits;
    if (p < H) { uint2 y = tf2x32((uint32_t)p,        (uint32_t)(p + H)); bits = y.x; }
    else       { uint2 y = tf2x32((uint32_t)(p - H),  (uint32_t)p);       bits = y.y; }
    uint32_t fb = (bits >> 9) | 0x3f800000u;          // [1,2)
    float u = __uint_as_float(fb) - 1.0f;             // [0,1)
    kv = ((unsigned long long)__float_as_uint(u) << 32) | (uint32_t)e;
  }
  keys[e] = kv;
  __syncthreads();

  // bitonic sort ascending, 256 elements / 256 threads
  for (int size = 2; size <= 256; size <<= 1) {
    for (int stride = size >> 1; stride > 0; stride >>= 1) {
      int partner = e ^ stride;
      unsigned long long a = keys[e], b2 = keys[partner];
      bool asc = ((e & size) == 0);
      unsigned long long lo = (a < b2) ? a : b2;
      unsigned long long hi = (a < b2) ? b2 : a;
      unsigned long long mine = ((e < partner) == asc) ? lo : hi;
      __syncthreads();
      keys[e] = mine;
      __syncthreads();
    }
  }

  if (e < K_) {
    int cand = (int)(keys[e] & 0xFFFFFFFFu);          // in [0, M-2]
    int self = n % M_;
    int nl = cand + (cand >= self);                   // skip-self remap
    neg_idx[(size_t)n * K_ + e] = (n / M_) * M_ + nl; // global label index
  }
}

// ---------------------------------------------------------------------------
// 4) per-batch similarity matrix S_b = T_hat_b x L_hat_b^T via f32 WMMA.
//    Each wave computes a 16x64 strip (4 accumulators), K-loop in steps of 4.
//    A frag (16x4 f32): lane -> M = lane%16, VGPR j -> K = 2*(lane/16)+j
//    B frag (4x16 f32): lane -> N = lane%16, VGPR j -> K = 2*(lane/16)+j
//    C/D (16x16 f32):   VGPR r, lane -> [M = r+8*(lane/16)][N = lane%16]
// ---------------------------------------------------------------------------
__global__ __launch_bounds__(256) void k_sim_wmma(const float* __restrict__ t_hat,
                                                  const float* __restrict__ l_hat,
                                                  float* __restrict__ S) {
  int wave = threadIdx.x >> 5;
  int lane = threadIdx.x & 31;
  int w  = blockIdx.x * 8 + wave;   // 0..1023 waves total
  int b  = w >> 6;                  // batch 0..15 (64 waves each)
  int rt = (w & 63) >> 2;           // row tile 0..15 (16 rows each)
  int cg = w & 3;                   // col group 0..3 (64 cols each)
  int lm = lane & 15;
  int lh = lane >> 4;

  const float* Arow = t_hat + ((size_t)(b * M_ + rt * 16 + lm)) * D_;
  const float* Bcol = l_hat + ((size_t)(b * M_ + cg * 64 + lm)) * D_;
  __builtin_prefetch(Arow, 0, 3);
  __builtin_prefetch(Bcol, 0, 3);

  v8f acc0 = {}, acc1 = {}, acc2 = {}, acc3 = {};
#pragma unroll 8
  for (int k = 0; k < D_; k += 4) {
    v2f a  = *(const v2f*)(Arow + k + 2 * lh);
    const float* Bb = Bcol + k + 2 * lh;
    v2f b0 = *(const v2f*)(Bb + 0 * 16 * D_);
    v2f b1 = *(const v2f*)(Bb + 1 * 16 * D_);
    v2f b2 = *(const v2f*)(Bb + 2 * 16 * D_);
    v2f b3 = *(const v2f*)(Bb + 3 * 16 * D_);
    acc0 = __builtin_amdgcn_wmma_f32_16x16x4_f32(false, a, false, b0, (short)0, acc0, false, false);
    acc1 = __builtin_amdgcn_wmma_f32_16x16x4_f32(false, a, false, b1, (short)0, acc1, false, false);
    acc2 = __builtin_amdgcn_wmma_f32_16x16x4_f32(false, a, false, b2, (short)0, acc2, false, false);
    acc3 = __builtin_amdgcn_wmma_f32_16x16x4_f32(false, a, false, b3, (short)0, acc3, false, false);
  }

  float* Sb = S + (size_t)b * M_ * M_;
#pragma unroll
  for (int r = 0; r < 8; ++r) {
    int row = rt * 16 + r + 8 * lh;
    size_t base = (size_t)row * M_ + cg * 64 + lm;
    Sb[base + 0 * 16] = acc0[r];
    Sb[base + 1 * 16] = acc1[r];
    Sb[base + 2 * 16] = acc2[r];
    Sb[base + 3 * 16] = acc3[r];
  }
}

// ---------------------------------------------------------------------------
// 5) per-target: -pos/T + logsumexp([self] + 100 negatives, scaled by 1/T)
// ---------------------------------------------------------------------------
__global__ void k_contrastive(const float* __restrict__ S,
                              const int* __restrict__ neg_idx,
                              float* __restrict__ outn) {
  __shared__ float red[128];
  int n = blockIdx.x;
  int t = threadIdx.x;
  int bb = n / M_, i = n % M_;
  const float* Srow = S + ((size_t)(bb * M_) + i) * M_;

  float v = -3.0e38f;
  if (t == 0) {
    v = Srow[i] * TEMP_INV;                          // k=0: the label itself
  } else if (t <= K_) {
    int g = neg_idx[(size_t)n * K_ + (t - 1)];
    v = Srow[g - bb * M_] * TEMP_INV;
  }
  red[t] = v; __syncthreads();
  for (int off = 64; off > 0; off >>= 1) { if (t < off) red[t] = fmaxf(red[t], red[t + off]); __syncthreads(); }
  float mx = red[0];
  __syncthreads();

  float ex = (t <= K_) ? __expf(v - mx) : 0.0f;
  red[t] = ex; __syncthreads();
  for (int off = 64; off > 0; off >>= 1) { if (t < off) red[t] += red[t + off]; __syncthreads(); }

  if (t == 0) {
    float lse  = mx + __logf(red[0]);
    float posv = Srow[i] * TEMP_INV;
    outn[n] = lse - posv;
  }
}

// ---------------------------------------------------------------------------
// 6) deterministic final reduction + diversity term
// ---------------------------------------------------------------------------
__global__ void k_final(const float* __restrict__ contr,
                        const float* __restrict__ perp,
                        float* __restrict__ out) {
  __shared__ float red[1024];
  int t = threadIdx.x;

  float s = 0.0f;
  for (int i = t; i < N_; i += 1024) s += contr[i];
  red[t] = s; __syncthreads();
  for (int off = 512; off > 0; off >>= 1) { if (t < off) red[t] += red[t + off]; __syncthreads(); }
  float cmean = red[0] / (float)N_;
  __syncthreads();

  float dsum = 0.0f;
  for (int i = t; i < G_ * V_; i += 1024) {
    float p = perp[i];
    dsum += p * __logf(fmaxf(p, 1e-9f));
  }
  red[t] = dsum; __syncthreads();
  for (int off = 512; off > 0; off >>= 1) { if (t < off) red[t] += red[t + off]; __syncthreads(); }

  if (t == 0) out[0] = cmean + ALPHA_ * (red[0] / (float)(G_ * V_));
}

// ---------------------------------------------------------------------------
extern "C" void kernel_launch(void* const* d_in, const int* in_sizes, int n_in,
                              void* d_out, int out_size, void* d_ws, size_t ws_size,
                              hipStream_t stream) {
  const float* enc   = (const float*)d_in[0];          // (B,T,D) f32
  const float* quant = (const float*)d_in[1];          // (B,T,D) f32
  const float* perp  = (const float*)d_in[2];          // (G,V)   f32
  const unsigned char* mask = (const unsigned char*)d_in[3]; // (B,T) bool

  char* ws = (char*)d_ws;
  float* t_hat = (float*)ws;  ws += (size_t)N_ * D_ * sizeof(float);   // 4 MB
  float* l_hat = (float*)ws;  ws += (size_t)N_ * D_ * sizeof(float);   // 4 MB
  float* S     = (float*)ws;  ws += (size_t)B_ * M_ * M_ * sizeof(float); // 4 MB
  int*   neg   = (int*)ws;    ws += (size_t)N_ * K_ * sizeof(int);     // 1.6 MB
  float* contr = (float*)ws;  ws += (size_t)N_ * sizeof(float);        // 16 KB
  int*   pos   = (int*)ws;    ws += (size_t)N_ * sizeof(int);          // 16 KB

  k_positions  <<<B_,  T_,  0, stream>>>(mask, pos);
  k_gather_norm<<<N_,  D_,  0, stream>>>(enc, quant, pos, t_hat, l_hat);
  k_negatives  <<<N_,  256, 0, stream>>>(neg);
  k_sim_wmma   <<<128, 256, 0, stream>>>(t_hat, l_hat, S);
  k_contrastive<<<N_,  128, 0, stream>>>(S, neg, contr);
  k_final      <<<1,   1024, 0, stream>>>(contr, perp, (float*)d_out);
}